// dMaSIFSiteEmbed_9826885174167
// MI455X (gfx1250) — compile-verified
//
#include <hip/hip_runtime.h>
#include <math.h>

// ---------------------------------------------------------------------------
// dMaSIF site embedding forward for MI455X (gfx1250, wave32).
// The dominant O(N^2) conv is mapped onto V_WMMA_F32_16X16X4_F32:
// per point i, tiles of 16 neighbors j form a (16j x 8c) H matrix that is
// multiplied by A2^T (8 x 64) as 4 N-tiles x 2 K-chunks of fp32 WMMA.
// fp32 WMMA keeps exact reference precision (reference is fp32 end to end).
// Occupancy: the j-loop is split across 4 waves per point (6400 waves total),
// partials reduced deterministically through LDS.
// ---------------------------------------------------------------------------

#define N_PTS  1600
#define IN_CH  16
#define EMB    64
#define NEGS   0.2f
#define GN_EPS 1e-5f

typedef float v2f __attribute__((ext_vector_type(2)));
typedef float v8f __attribute__((ext_vector_type(8)));

__device__ __forceinline__ float leaky(float x) { return x >= 0.f ? x : NEGS * x; }

// ---------------- orientation scores: w[i] ----------------
__global__ void orient_kernel(const float* __restrict__ feats,
                              const float* __restrict__ w1, const float* __restrict__ b1,
                              const float* __restrict__ w2, const float* __restrict__ b2,
                              float* __restrict__ wout) {
  int i = blockIdx.x * blockDim.x + threadIdx.x;
  if (i >= N_PTS) return;
  float x[IN_CH];
  #pragma unroll
  for (int c = 0; c < IN_CH; ++c) x[c] = feats[i * IN_CH + c];
  float acc = b2[0];
  #pragma unroll
  for (int u = 0; u < 16; ++u) {
    float s = b1[u];
    #pragma unroll
    for (int c = 0; c < IN_CH; ++c) s += w1[u * IN_CH + c] * x[c];
    acc += w2[u] * leaky(s);
  }
  wout[i] = acc;
}

// ---------------- build_nuv: one wave per point i ----------------
__global__ __launch_bounds__(256) void nuv_kernel(const float* __restrict__ xyz,
                                                  const float* __restrict__ nrm,
                                                  const float* __restrict__ w,
                                                  float* __restrict__ nuv) {
  const int lane = threadIdx.x & 31;
  const int wave = threadIdx.x >> 5;
  const int i = blockIdx.x * 8 + wave;
  const float inv_r = 1.0f / 9.0f;
  const float pix = xyz[i * 3 + 0] * inv_r;
  const float piy = xyz[i * 3 + 1] * inv_r;
  const float piz = xyz[i * 3 + 2] * inv_r;
  const float nx = nrm[i * 3 + 0], ny = nrm[i * 3 + 1], nz = nrm[i * 3 + 2];
  const float s = (nz >= 0.f) ? 1.f : -1.f;
  const float a = -1.f / (s + nz);
  const float b = nx * ny * a;
  const float ux = 1.f + s * nx * nx * a, uy = s * b, uz = -s * nx;
  const float vx = b, vy = s + ny * ny * a, vz = -ny;

  float ov0 = 0.f, ov1 = 0.f;
  for (int j = lane; j < N_PTS; j += 32) {
    float dx = xyz[j * 3 + 0] * inv_r - pix;
    float dy = xyz[j * 3 + 1] * inv_r - piy;
    float dz = xyz[j * 3 + 2] * inv_r - piz;
    float win = __expf(-(dx * dx + dy * dy + dz * dz));
    float ww = win * w[j];
    ov0 += ww * (ux * dx + uy * dy + uz * dz);
    ov1 += ww * (vx * dx + vy * dy + vz * dz);
  }
  #pragma unroll
  for (int off = 16; off >= 1; off >>= 1) {
    ov0 += __shfl_xor(ov0, off, 32);
    ov1 += __shfl_xor(ov1, off, 32);
  }
  if (lane == 0) {
    float nn = fmaxf(sqrtf(ov0 * ov0 + ov1 * ov1), 1e-12f);
    float ex = ov0 / nn, ey = ov1 / nn;
    float* o = nuv + i * 9;
    o[0] = nx; o[1] = ny; o[2] = nz;
    o[3] = ex * ux + ey * vx; o[4] = ex * uy + ey * vy; o[5] = ex * uz + ey * vz;
    o[6] = -ey * ux + ex * vx; o[7] = -ey * uy + ex * vy; o[8] = -ey * uz + ex * vz;
  }
}

// ---------------- fused 2-layer MLP, leaky_relu both stages ----------------
__global__ __launch_bounds__(EMB) void mlp2_leaky_kernel(const float* __restrict__ x, int in_d,
                                                         const float* __restrict__ W1, const float* __restrict__ b1,
                                                         const float* __restrict__ W2, const float* __restrict__ b2,
                                                         float* __restrict__ out) {
  __shared__ float t1[EMB];
  const int i = blockIdx.x;
  const int c = threadIdx.x;
  const float* xr = x + i * in_d;
  float sacc = b1[c];
  for (int d = 0; d < in_d; ++d) sacc += W1[c * in_d + d] * xr[d];
  t1[c] = leaky(sacc);
  __syncthreads();
  float s2 = b2[c];
  #pragma unroll 8
  for (int d = 0; d < EMB; ++d) s2 += W2[c * EMB + d] * t1[d];
  out[i * EMB + c] = leaky(s2);
}

// ---------------- GroupNorm global reduce (mean over all n & channels/group) ----
__global__ __launch_bounds__(256) void gn_reduce_kernel(const float* __restrict__ x,
                                                        float* __restrict__ stats) {
  const int g = blockIdx.x;           // 4 groups
  const int tid = threadIdx.x;        // 256 threads
  float s = 0.f, sq = 0.f;
  for (int idx = tid; idx < N_PTS * 16; idx += 256) {
    int i = idx >> 4;
    int c = (g << 4) + (idx & 15);
    float v = x[i * EMB + c];
    s += v; sq += v * v;
  }
  __shared__ float ss[256], qq[256];
  ss[tid] = s; qq[tid] = sq;
  __syncthreads();
  for (int off = 128; off > 0; off >>= 1) {
    if (tid < off) { ss[tid] += ss[tid + off]; qq[tid] += qq[tid + off]; }
    __syncthreads();
  }
  if (tid == 0) {
    const float invN = 1.0f / (float)(N_PTS * 16);
    float mean = ss[0] * invN;
    float var = qq[0] * invN - mean * mean;
    stats[g * 2 + 0] = mean;
    stats[g * 2 + 1] = rsqrtf(fmaxf(var, 0.f) + GN_EPS);
  }
}

__global__ void gn_apply_kernel(float* __restrict__ x, const float* __restrict__ stats,
                                const float* __restrict__ gamma, const float* __restrict__ beta) {
  int idx = blockIdx.x * blockDim.x + threadIdx.x;
  if (idx >= N_PTS * EMB) return;
  int c = idx & 63;
  int g = c >> 4;
  float v = (x[idx] - stats[2 * g]) * stats[2 * g + 1];
  x[idx] = v * gamma[c] + beta[c];
}

// ---------------- core quasi-geodesic conv with fp32 WMMA --------------------
// 8 waves/block = 2 points x 4 j-splits. For each tile of 16 neighbors j:
//   lane L computes window & h[0..7] for j = jb + (L&15)   (lanes 16-31 dup)
//   A-frag (16x4 f32): lanes 0-15 hold K={0,1}, lanes 16-31 K={2,3}
//   B-frag (4x16, = A2^T) mirrored: vgpr0 K = ch*4 + (L<16?0:2), vgpr1 K+1
//   D (16x16): vgpr r holds row m=r (L<16) / m=r+8 (L>=16), col n = L&15
// Partial out[i,0:64] per split reduced via LDS (deterministic order).
__global__ __launch_bounds__(256) void conv_wmma_kernel(
    const float* __restrict__ xyz, const float* __restrict__ nrm,
    const float* __restrict__ nuv, const float* __restrict__ f,
    const float* __restrict__ A1, const float* __restrict__ B1,
    const float* __restrict__ A2, const float* __restrict__ B2,
    float* __restrict__ out) {
  const int lane = threadIdx.x & 31;
  const int wave = threadIdx.x >> 5;   // 0..7
  const int iloc = wave >> 2;          // 0..1 : point within block
  const int split = wave & 3;          // 0..3 : j-range split
  const int i = blockIdx.x * 2 + iloc;
  const float scale = 0.07856742013183862f;  // 1/(sqrt(2)*9)

  const float pix = xyz[i * 3 + 0] * scale;
  const float piy = xyz[i * 3 + 1] * scale;
  const float piz = xyz[i * 3 + 2] * scale;
  float nv[9];
  #pragma unroll
  for (int k = 0; k < 9; ++k) nv[k] = nuv[i * 9 + k];

  float a1r[24], b1r[8];
  #pragma unroll
  for (int k = 0; k < 24; ++k) a1r[k] = A1[k];
  #pragma unroll
  for (int k = 0; k < 8; ++k) b1r[k] = B1[k];

  const int hcol = lane & 15;
  const int kb = (lane & 16) ? 2 : 0;    // K half for this lane group
  const int mhalf = (lane & 16) >> 1;    // row offset 0 or 8 for D rows

  v2f bfrag[4][2];
  float b2t[4];
  #pragma unroll
  for (int t = 0; t < 4; ++t) {
    const int h = t * 16 + hcol;
    #pragma unroll
    for (int ch = 0; ch < 2; ++ch) {
      bfrag[t][ch].x = A2[h * 8 + ch * 4 + kb + 0];
      bfrag[t][ch].y = A2[h * 8 + ch * 4 + kb + 1];
    }
    b2t[t] = B2[h];
  }

  float acc[4] = {0.f, 0.f, 0.f, 0.f};
  const bool hi = (lane & 16) != 0;

  for (int jb = split * 16; jb < N_PTS; jb += 64) {
    const int j = jb + hcol;
    float dx = xyz[j * 3 + 0] * scale - pix;
    float dy = xyz[j * 3 + 1] * scale - piy;
    float dz = xyz[j * 3 + 2] * scale - piz;
    float njx = nrm[j * 3 + 0], njy = nrm[j * 3 + 1], njz = nrm[j * 3 + 2];
    float t2 = 2.f - (nv[0] * njx + nv[1] * njy + nv[2] * njz);
    float win = __expf(-(dx * dx + dy * dy + dz * dz) * t2 * t2);
    float X0 = nv[0] * dx + nv[1] * dy + nv[2] * dz;
    float X1 = nv[3] * dx + nv[4] * dy + nv[5] * dz;
    float X2 = nv[6] * dx + nv[7] * dy + nv[8] * dz;
    float h[8];
    #pragma unroll
    for (int c = 0; c < 8; ++c)
      h[c] = fmaxf(a1r[c * 3 + 0] * X0 + a1r[c * 3 + 1] * X1 + a1r[c * 3 + 2] * X2 + b1r[c], 0.f);

    float winm[8];
    #pragma unroll
    for (int r = 0; r < 8; ++r) winm[r] = __shfl(win, r + mhalf, 32);

    v2f af0, af1;
    af0.x = hi ? h[2] : h[0];
    af0.y = hi ? h[3] : h[1];
    af1.x = hi ? h[6] : h[4];
    af1.y = hi ? h[7] : h[5];

    const float* frow = f + (jb + mhalf) * EMB;
    #pragma unroll
    for (int t = 0; t < 4; ++t) {
      v8f d = {};
      d = __builtin_amdgcn_wmma_f32_16x16x4_f32(false, af0, false, bfrag[t][0], (short)0, d, false, false);
      d = __builtin_amdgcn_wmma_f32_16x16x4_f32(false, af1, false, bfrag[t][1], (short)0, d, false, false);
      float sacc = 0.f;
      #pragma unroll
      for (int r = 0; r < 8; ++r) {
        float g = fmaxf(d[r] + b2t[t], 0.f);
        sacc += winm[r] * g * frow[r * EMB + t * 16 + hcol];
      }
      acc[t] += sacc;
    }
  }

  // fold the two row-halves (lane L and L+16 share the same h column)
  #pragma unroll
  for (int t = 0; t < 4; ++t) acc[t] += __shfl_xor(acc[t], 16, 32);

  // deterministic cross-split reduction through LDS
  __shared__ float red[2][4][EMB];
  if (lane < 16) {
    #pragma unroll
    for (int t = 0; t < 4; ++t) red[iloc][split][t * 16 + lane] = acc[t];
  }
  __syncthreads();
  if (split == 0) {
    for (int c = lane; c < EMB; c += 32) {
      float sfin = red[iloc][0][c] + red[iloc][1][c] + red[iloc][2][c] + red[iloc][3][c];
      out[i * EMB + c] = sfin;
    }
  }
}

// ---------------- residual: x_out = x_in@tw^T + tb + mlp(relu)(xi) ----------
__global__ __launch_bounds__(EMB) void resid_kernel(const float* __restrict__ x_in, int in_d,
                                                    const float* __restrict__ xi,
                                                    const float* __restrict__ mw1, const float* __restrict__ mb1,
                                                    const float* __restrict__ mw2, const float* __restrict__ mb2,
                                                    const float* __restrict__ tw, const float* __restrict__ tb,
                                                    float* __restrict__ x_out) {
  __shared__ float t1[EMB];
  const int i = blockIdx.x;
  const int c = threadIdx.x;
  const float* xr = xi + i * EMB;
  float s = mb1[c];
  #pragma unroll 8
  for (int d = 0; d < EMB; ++d) s += mw1[c * EMB + d] * xr[d];
  t1[c] = fmaxf(s, 0.f);
  __syncthreads();
  float v = mb2[c];
  #pragma unroll 8
  for (int d = 0; d < EMB; ++d) v += mw2[c * EMB + d] * t1[d];
  float tv = tb[c];
  const float* xinr = x_in + i * in_d;
  for (int d = 0; d < in_d; ++d) tv += tw[c * in_d + d] * xinr[d];
  x_out[i * EMB + c] = tv + v;
}

// ---------------------------------------------------------------------------
extern "C" void kernel_launch(void* const* d_in, const int* in_sizes, int n_in,
                              void* d_out, int out_size, void* d_ws, size_t ws_size,
                              hipStream_t stream) {
  const float* xyz = (const float*)d_in[0];
  const float* nrm = (const float*)d_in[1];
  const float* feats = (const float*)d_in[2];
  const float* ow1 = (const float*)d_in[3];
  const float* ob1 = (const float*)d_in[4];
  const float* ow2 = (const float*)d_in[5];
  const float* ob2 = (const float*)d_in[6];

  float* ws = (float*)d_ws;
  float* wbuf  = ws;                       // 1600
  float* nuv   = ws + 1600;                // 14400
  float* bufA  = ws + 16000;               // 102400 (f)
  float* bufB  = bufA + N_PTS * EMB;       // 102400 (conv raw)
  float* bufC  = bufB + N_PTS * EMB;       // 102400 (post MLP/GN)
  float* x1    = bufC + N_PTS * EMB;       // 102400 (layer0 output)
  float* stats = x1 + N_PTS * EMB;         // 8

  orient_kernel<<<(N_PTS + 255) / 256, 256, 0, stream>>>(feats, ow1, ob1, ow2, ob2, wbuf);
  nuv_kernel<<<N_PTS / 8, 256, 0, stream>>>(xyz, nrm, wbuf, nuv);

  const float* x_cur = feats;
  int in_d = IN_CH;
  for (int l = 0; l < 2; ++l) {
    const int base = 7 + l * 22;  // params flattened in setup_inputs() insertion order
    const float* wi1 = (const float*)d_in[base + 0];
    const float* bi1 = (const float*)d_in[base + 1];
    const float* wi2 = (const float*)d_in[base + 2];
    const float* bi2 = (const float*)d_in[base + 3];
    const float* gi  = (const float*)d_in[base + 4];
    const float* gbi = (const float*)d_in[base + 5];
    const float* A1  = (const float*)d_in[base + 6];
    const float* B1  = (const float*)d_in[base + 7];
    const float* A2  = (const float*)d_in[base + 8];
    const float* B2  = (const float*)d_in[base + 9];
    const float* wo1 = (const float*)d_in[base + 10];
    const float* bo1 = (const float*)d_in[base + 11];
    const float* wo2 = (const float*)d_in[base + 12];
    const float* bo2 = (const float*)d_in[base + 13];
    const float* go  = (const float*)d_in[base + 14];
    const float* gbo = (const float*)d_in[base + 15];
    const float* mw1 = (const float*)d_in[base + 16];
    const float* mb1 = (const float*)d_in[base + 17];
    const float* mw2 = (const float*)d_in[base + 18];
    const float* mb2 = (const float*)d_in[base + 19];
    const float* tw  = (const float*)d_in[base + 20];
    const float* tb  = (const float*)d_in[base + 21];

    mlp2_leaky_kernel<<<N_PTS, EMB, 0, stream>>>(x_cur, in_d, wi1, bi1, wi2, bi2, bufA);
    gn_reduce_kernel<<<4, 256, 0, stream>>>(bufA, stats);
    gn_apply_kernel<<<(N_PTS * EMB + 255) / 256, 256, 0, stream>>>(bufA, stats, gi, gbi);

    conv_wmma_kernel<<<N_PTS / 2, 256, 0, stream>>>(xyz, nrm, nuv, bufA, A1, B1, A2, B2, bufB);

    mlp2_leaky_kernel<<<N_PTS, EMB, 0, stream>>>(bufB, EMB, wo1, bo1, wo2, bo2, bufC);
    gn_reduce_kernel<<<4, 256, 0, stream>>>(bufC, stats);
    gn_apply_kernel<<<(N_PTS * EMB + 255) / 256, 256, 0, stream>>>(bufC, stats, go, gbo);

    float* x_next = (l == 0) ? x1 : (float*)d_out;
    resid_kernel<<<N_PTS, EMB, 0, stream>>>(x_cur, in_d, bufC, mw1, mb1, mw2, mb2, tw, tb, x_next);
    x_cur = x_next;
    in_d = EMB;
  }
}